// SOLD2_detector_69466801045725
// MI455X (gfx1250) — compile-verified
//
#include <hip/hip_runtime.h>
#include <hip/hip_bf16.h>
#include <stdint.h>

#define NJ     512
#define HH     512
#define WW     512
#define NPIX   (HH*WW)
#define NPAIRS ((NJ*(NJ+1))/2)   // 131328 (i<=j)
#define NBIN   4096

// ---- workspace layout (32-bit words) ----
#define WS_CNT1 0
#define WS_SUM1 4096
#define WS_CNT2 8192
#define WS_SUM2 12288
#define WS_SC   16384
#define WS_IMG_BYTES 69632        // u8 refined heatmap, 262144 bytes (16B aligned)

#if defined(__AMDGCN__) && defined(__has_builtin)
#if __has_builtin(__builtin_amdgcn_global_load_async_to_lds_b128)
#define HAVE_ASYNC_LDS 1
#endif
#endif

#ifdef HAVE_ASYNC_LDS
typedef int v4i_vs __attribute__((vector_size(16)));
typedef __attribute__((address_space(1))) v4i_vs* async_gptr_t;
typedef __attribute__((address_space(3))) v4i_vs* async_lptr_t;
#endif

// ---------------- stage 0: zero histogram workspace ----------------
__global__ void k_init(uint32_t* __restrict__ ws) {
  int n = WS_SC + 8;
  for (int i = blockIdx.x * blockDim.x + threadIdx.x; i < n; i += gridDim.x * blockDim.x)
    ws[i] = 0u;
}

// ---------------- stage 1: coarse histogram of valid values ----------------
__global__ void k_hist1(const float* __restrict__ hm, uint32_t* __restrict__ ws) {
  __shared__ uint32_t lc[NBIN];
  __shared__ float    ls[NBIN];
  for (int b = threadIdx.x; b < NBIN; b += blockDim.x) { lc[b] = 0u; ls[b] = 0.0f; }
  __syncthreads();
  for (int i = blockIdx.x * blockDim.x + threadIdx.x; i < NPIX; i += gridDim.x * blockDim.x) {
    float v = hm[i];
    if (v > 0.002f) {
      int b = (int)(v * (float)NBIN);
      b = b < 0 ? 0 : (b > NBIN - 1 ? NBIN - 1 : b);
      atomicAdd(&lc[b], 1u);
      atomicAdd(&ls[b], v);
    }
  }
  __syncthreads();
  uint32_t* cnt1 = ws + WS_CNT1;
  float*    sum1 = (float*)(ws + WS_SUM1);
  for (int b = threadIdx.x; b < NBIN; b += blockDim.x) {
    if (lc[b]) { atomicAdd(&cnt1[b], lc[b]); atomicAdd(&sum1[b], ls[b]); }
  }
}

// ---------------- stage 2: find straddling bin ----------------
__global__ void k_select1(uint32_t* __restrict__ ws) {
  if (threadIdx.x != 0 || blockIdx.x != 0) return;
  uint32_t* cnt1 = ws + WS_CNT1;
  float*    sum1 = (float*)(ws + WS_SUM1);
  uint32_t nv = 0;
  for (int b = 0; b < NBIN; ++b) nv += cnt1[b];
  int k = (int)ceilf((float)nv * 0.2f);
  if (k < 1) k = 1;
  long long cum = 0; double sumAbove = 0.0; int b;
  for (b = NBIN - 1; b > 0; --b) {
    if (cum + (long long)cnt1[b] >= (long long)k) break;
    cum += (long long)cnt1[b];
    sumAbove += (double)sum1[b];
  }
  int*   sci = (int*)(ws + WS_SC);
  float* scf = (float*)(ws + WS_SC);
  sci[0] = b;                      // threshold bin
  scf[1] = (float)sumAbove;        // sum strictly above bin
  sci[2] = (int)((long long)k - cum); // remaining elements inside bin
  sci[3] = k;
}

// ---------------- stage 3: fine histogram inside straddling bin ----------------
__global__ void k_hist2(const float* __restrict__ hm, uint32_t* __restrict__ ws) {
  __shared__ uint32_t lc[NBIN];
  __shared__ float    ls[NBIN];
  int tb = ((const int*)(ws + WS_SC))[0];
  for (int b = threadIdx.x; b < NBIN; b += blockDim.x) { lc[b] = 0u; ls[b] = 0.0f; }
  __syncthreads();
  for (int i = blockIdx.x * blockDim.x + threadIdx.x; i < NPIX; i += gridDim.x * blockDim.x) {
    float v = hm[i];
    if (v > 0.002f) {
      int b = (int)(v * (float)NBIN);
      b = b < 0 ? 0 : (b > NBIN - 1 ? NBIN - 1 : b);
      if (b == tb) {
        int sb = (int)((v * (float)NBIN - (float)tb) * (float)NBIN);
        sb = sb < 0 ? 0 : (sb > NBIN - 1 ? NBIN - 1 : sb);
        atomicAdd(&lc[sb], 1u);
        atomicAdd(&ls[sb], v);
      }
    }
  }
  __syncthreads();
  uint32_t* cnt2 = ws + WS_CNT2;
  float*    sum2 = (float*)(ws + WS_SUM2);
  for (int b = threadIdx.x; b < NBIN; b += blockDim.x) {
    if (lc[b]) { atomicAdd(&cnt2[b], lc[b]); atomicAdd(&sum2[b], ls[b]); }
  }
}

// ---------------- stage 4: finish top-k mean ----------------
__global__ void k_select2(uint32_t* __restrict__ ws) {
  if (threadIdx.x != 0 || blockIdx.x != 0) return;
  int*   sci = (int*)(ws + WS_SC);
  float* scf = (float*)(ws + WS_SC);
  int rem = sci[2];
  int k   = sci[3];
  double total = (double)scf[1];
  uint32_t* cnt2 = ws + WS_CNT2;
  float*    sum2 = (float*)(ws + WS_SUM2);
  long long cum = 0;
  for (int sb = NBIN - 1; sb >= 0; --sb) {
    uint32_t c = cnt2[sb];
    if (cum + (long long)c >= (long long)rem) {
      long long need = (long long)rem - cum;
      float avg = c ? (sum2[sb] / (float)c) : 0.0f;
      total += (double)need * (double)avg;
      break;
    }
    cum += (long long)c;
    total += (double)sum2[sb];
  }
  scf[4] = (float)(total / (double)k);   // max20
}

// ---------------- stage 5: refine heatmap, write f32 out + u8 image ----------------
__global__ void k_apply(const float* __restrict__ hm, uint32_t* __restrict__ ws,
                        float* __restrict__ out_hm) {
  float max20 = ((const float*)(ws + WS_SC))[4];
  unsigned char* img = (unsigned char*)ws + WS_IMG_BYTES;
  for (int i = blockIdx.x * blockDim.x + threadIdx.x; i < NPIX; i += gridDim.x * blockDim.x) {
    float v = hm[i] / max20;
    v = fminf(fmaxf(v, 0.0f), 1.0f);
    out_hm[i] = v;
    img[i] = (unsigned char)(int)rintf(v * 255.0f);
  }
}

// ---------------- stage 6: pairwise segment scoring with LDS-resident image ----------------
__global__ __launch_bounds__(1024, 1)
void k_pairs(const float* __restrict__ junc, const uint32_t* __restrict__ ws,
             float* __restrict__ out) {
  __shared__ __align__(16) unsigned char s_hm[NPIX];   // 256 KB u8 image in LDS (CDNA5: 320 KB/WGP)
  const unsigned char* img = (const unsigned char*)ws + WS_IMG_BYTES;

  // Cooperative 256 KB global -> LDS stage (async b128 copies, ASYNCcnt-tracked)
  const int nchunk = NPIX / 16;  // 16384 x 16B
#ifdef HAVE_ASYNC_LDS
  for (int c = threadIdx.x; c < nchunk; c += blockDim.x) {
    int off = c * 16;
    __builtin_amdgcn_global_load_async_to_lds_b128(
        (async_gptr_t)(img + off),
        (async_lptr_t)(&s_hm[off]),
        0, 0);
  }
  asm volatile("s_wait_asynccnt 0" ::: "memory");
#else
  for (int c = threadIdx.x; c < nchunk; c += blockDim.x) {
    int off = c * 16;
    *(uint4*)(s_hm + off) = *(const uint4*)(img + off);
  }
#endif
  __syncthreads();

  int p = blockIdx.x * blockDim.x + threadIdx.x;
  if (p >= NPAIRS) return;

  // decode upper-triangular (i<=j) pair index
  int i = (int)(512.0f - sqrtf(fmaxf(262144.0f - 2.0f * (float)p, 0.0f)));
  i = i < 0 ? 0 : (i > 511 ? 511 : i);
  while (i > 0 && (i * (1025 - i)) / 2 > p) --i;
  while (((i + 1) * (1024 - i)) / 2 <= p) ++i;
  int j = i + (p - (i * (1025 - i)) / 2);

  float hi = junc[2 * i], wi = junc[2 * i + 1];
  float hj = junc[2 * j], wj = junc[2 * j + 1];
  float dh = hi - hj, dw = wi - wj;
  float seg = sqrtf(dh * dh + dw * dw);
  float th  = 0.70710678f + 2.0f * (seg / 724.0773439f);  // 0.5*sqrt(2) + lambda*seg/sqrt(H^2+W^2)
  float th2 = th * th;

  unsigned sum_q = 0;
  int cnt_in = 0;
  for (int s = 0; s < 64; ++s) {
    float t  = (float)s * (1.0f / 63.0f);
    float ch = hi * t + hj * (1.0f - t);
    float cw = wi * t + wj * (1.0f - t);
    ch = fminf(fmaxf(ch, 0.0f), 511.0f);
    cw = fminf(fmaxf(cw, 0.0f), 511.0f);
    float rch = rintf(ch), rcw = rintf(cw);   // matches jnp.round (half-to-even)

    int   rowb[7], colb[7];
    float dy2[7], dx2[7];
#pragma unroll
    for (int o = 0; o < 7; ++o) {
      float ph = fminf(fmaxf(rch + (float)(o - 3), 0.0f), 511.0f);
      float pw = fminf(fmaxf(rcw + (float)(o - 3), 0.0f), 511.0f);
      float dy = ph - ch, dx = pw - cw;
      dy2[o]  = dy * dy;
      dx2[o]  = dx * dx;
      rowb[o] = ((int)ph) << 9;
      colb[o] = (int)pw;
    }

    unsigned best = 0;
#pragma unroll
    for (int oy = 0; oy < 7; ++oy) {
      float dyv = dy2[oy];
      int   rb  = rowb[oy];
#pragma unroll
      for (int ox = 0; ox < 7; ++ox) {
        unsigned v = (unsigned)s_hm[rb + colb[ox]];   // ds_load_u8 from LDS
        v = ((dyv + dx2[ox]) < th2) ? v : 0u;
        best = v > best ? v : best;
      }
    }
    sum_q += best;
    cnt_in += (best >= 128u) ? 1 : 0;   // q/255 > 0.5
  }

  float mean = (float)sum_q * (1.0f / (64.0f * 255.0f));
  // detection: mean>0.5 AND inlier ratio >= 0.9 (cnt/64>=0.9 <=> cnt>=58), triu k=1 kills diagonal
  int det = (mean > 0.5f) && (cnt_in >= 58) && (i != j);

  float* lm = out;           // line_map   (int32 in ref; stored as 0.0/1.0 floats)
  float* ms = out + NPIX;    // mean_scores
  float dv = det ? 1.0f : 0.0f;
  ms[i * 512 + j] = mean; ms[j * 512 + i] = mean;
  lm[i * 512 + j] = dv;   lm[j * 512 + i] = dv;
}

extern "C" void kernel_launch(void* const* d_in, const int* in_sizes, int n_in,
                              void* d_out, int out_size, void* d_ws, size_t ws_size,
                              hipStream_t stream) {
  (void)in_sizes; (void)n_in; (void)out_size; (void)ws_size;
  const float* junc = (const float*)d_in[0];   // (512,2) f32
  const float* hm   = (const float*)d_in[1];   // (512,512) f32
  float*    out = (float*)d_out;               // [line_map | mean_scores | hm] each 512*512
  uint32_t* ws  = (uint32_t*)d_ws;

  k_init   <<<64,  256, 0, stream>>>(ws);
  k_hist1  <<<256, 256, 0, stream>>>(hm, ws);
  k_select1<<<1,   1,   0, stream>>>(ws);
  k_hist2  <<<256, 256, 0, stream>>>(hm, ws);
  k_select2<<<1,   1,   0, stream>>>(ws);
  k_apply  <<<256, 256, 0, stream>>>(hm, ws, out + 2 * NPIX);

  int blocks = (NPAIRS + 1023) / 1024;   // 129 blocks x 1024 threads, 1 pair/thread
  k_pairs  <<<blocks, 1024, 0, stream>>>(junc, ws, out);
}